// BiQRNNAtt_1022202216455
// MI455X (gfx1250) — compile-verified
//
#include <hip/hip_runtime.h>
#include <hip/hip_bf16.h>
#include <math.h>

typedef __attribute__((ext_vector_type(2))) float v2f;
typedef __attribute__((ext_vector_type(8))) float v8f;

#define NTOK   (32 * 4096)          // B*S = 131072 token rows
#define EDIM   256
#define HDIM   256
#define SEQ    4096
#define BATCH  32
#define LDS_STRIDE 260              // 256 + 4 pad -> conflict-free b64 LDS reads

static __device__ __forceinline__ float fast_rcp(float x) {
    return __builtin_amdgcn_rcpf(x);        // v_rcp_f32
}
// tanh(x) = 1 - 2/(1+exp(2x)); saturates correctly: exp->inf => rcp->0 => +1
static __device__ __forceinline__ float fast_tanh(float x) {
    return 1.0f - 2.0f * fast_rcp(1.0f + __expf(2.0f * x));
}
// sigmoid(x) = 1/(1+exp(-x))
static __device__ __forceinline__ float fast_sigmoid(float x) {
    return fast_rcp(1.0f + __expf(-x));
}

// ---------------------------------------------------------------------------
// Kernel 1: fused embedding gather + GEMM (f32 WMMA 16x16x4) + bias + gate
// activations. Computes z/f gates for BOTH directions (o-gate skipped: dead).
// Each workgroup: 64 token rows staged in LDS; 8 waves x 8 N-tiles each;
// each wave keeps 4 M-tile accumulators so the B fragment is reused 4x.
// dir/gate/destination are wave-uniform (derived from the scalar tile index)
// so the epilogue select is scalar, not exec-mask divergence.
// ---------------------------------------------------------------------------
__global__ __launch_bounds__(256)
void gemm_gates_kernel(const int* __restrict__ X,
                       const float* __restrict__ emb,
                       const float* __restrict__ wf, const float* __restrict__ bfv,
                       const float* __restrict__ wb, const float* __restrict__ bbv,
                       float* __restrict__ gateZ, float* __restrict__ gateF) {
    __shared__ float As[64 * LDS_STRIDE];   // 64 rows x 256 f32 (padded) ~65 KB

    const int tid = threadIdx.x;
    const int wg  = blockIdx.x;             // 2048 workgroups x 64 rows

    // ---- stage A: gather 64 embedding rows into LDS (float4 moves) ----
    {
        const int r = tid >> 2;             // row 0..63
        const int q = tid & 3;              // quarter of the 256-wide row
        const int row_global = wg * 64 + r;
        const int token = X[row_global];
        const float4* src = (const float4*)(emb + (size_t)token * EDIM + q * 64);
        float* dst = &As[r * LDS_STRIDE + q * 64];
#pragma unroll
        for (int j = 0; j < 16; ++j) {
            float4 v = src[j];
            *(float4*)(dst + j * 4) = v;
        }
    }
    __syncthreads();

    const int wave = tid >> 5;
    const int lane = tid & 31;
    const int m    = lane & 15;             // A row within tile / output column low bits
    const int hh   = lane >> 4;             // half-wave select (K pair)

    for (int j = 0; j < 8; ++j) {
        const int nt   = j * 8 + wave;      // N-tile 0..63 (1024 live channels)
        // scalar (wave-uniform) decode of the 16-wide channel tile:
        const int dir  = nt >> 5;           // 0 = fwd conv, 1 = bwd conv
        const int gate = (nt >> 4) & 1;     // 0 = z (tanh), 1 = f (sigmoid)
        const int within_base = (nt & 31) * 16;   // conv-weight row base (0..511)
        const int c_base      = (nt & 15) * 16;   // hidden-channel base  (0..255)

        const int within = within_base + (lane & 15);
        const int c      = c_base      + (lane & 15);

        const float* W    = dir ? wb  : wf;
        const float* Bv   = dir ? bbv : bfv;
        const float* wrow = W + (size_t)within * EDIM;
        const float  bias = Bv[within];

        v8f acc[4];
#pragma unroll
        for (int mt = 0; mt < 4; ++mt) acc[mt] = (v8f){0,0,0,0,0,0,0,0};

        for (int k0 = 0; k0 < EDIM; k0 += 4) {
            const int ks = k0 + 2 * hh;     // this lane's K pair
            const v2f bfrag = *(const v2f*)(wrow + ks);
#pragma unroll
            for (int mt = 0; mt < 4; ++mt) {
                const v2f afrag = *(const v2f*)&As[(mt * 16 + m) * LDS_STRIDE + ks];
                acc[mt] = __builtin_amdgcn_wmma_f32_16x16x4_f32(
                    /*neg_a=*/false, afrag, /*neg_b=*/false, bfrag,
                    /*c_mod=*/(short)0, acc[mt], /*reuse_a=*/false, /*reuse_b=*/false);
            }
        }

        // epilogue: bias + activation + scatter to gate arrays [dir][token][c]
        float* dst = (gate == 0 ? gateZ : gateF) + (size_t)dir * NTOK * HDIM;
#pragma unroll
        for (int mt = 0; mt < 4; ++mt) {
#pragma unroll
            for (int e = 0; e < 8; ++e) {
                const float x = acc[mt][e] + bias;
                const float y = (gate == 0) ? fast_tanh(x) : fast_sigmoid(x);
                const int tok = wg * 64 + mt * 16 + e + 8 * hh;  // C layout: M=e(+8)
                dst[(size_t)tok * HDIM + c] = y;
            }
        }
    }
}

// ---------------------------------------------------------------------------
// Kernel 2: fo-pool recurrence h_t = f*h_{t-1} + (1-f)*z, per (dir,b,c).
// Backward dir reads time-reversed, writes in processing order (NOT re-flipped)
// exactly as the reference. Loads are independent of the recurrence -> pipeline.
// ---------------------------------------------------------------------------
__global__ __launch_bounds__(256)
void scan_kernel(const float* __restrict__ gateZ, const float* __restrict__ gateF,
                 float* __restrict__ h) {
    const int idx = blockIdx.x * 256 + threadIdx.x;   // 16384 total
    const int dir = idx >> 13;
    const int rem = idx & 8191;
    const int b   = rem >> 8;
    const int c   = rem & 255;

    const float* Z = gateZ + (size_t)dir * NTOK * HDIM;
    const float* F = gateF + (size_t)dir * NTOK * HDIM;
    const size_t base = (size_t)b * SEQ;
    float hv = 0.0f;

#pragma unroll 4
    for (int i = 0; i < SEQ; ++i) {
        const int t = dir ? (SEQ - 1 - i) : i;
        const float z = Z[(base + t) * HDIM + c];
        const float f = F[(base + t) * HDIM + c];
        hv = f * hv + (1.0f - f) * z;
        h[(base + i) * (2 * HDIM) + dir * HDIM + c] = hv;
    }
}

// ---------------------------------------------------------------------------
// Kernel 3: attention logits m[b,s] = dot(h[b,s,:], Mu).  One wave per row.
// ---------------------------------------------------------------------------
__global__ __launch_bounds__(256)
void logits_kernel(const float* __restrict__ h, const float* __restrict__ Mu,
                   float* __restrict__ logits) {
    const int wave = threadIdx.x >> 5;
    const int lane = threadIdx.x & 31;
    const int row  = blockIdx.x * 8 + wave;           // 131072 rows
    const float* hr = h + (size_t)row * (2 * HDIM);
    float s = 0.0f;
#pragma unroll
    for (int k = 0; k < 16; ++k) {
        const int c = lane + k * 32;                  // coalesced
        s += hr[c] * Mu[c];
    }
#pragma unroll
    for (int off = 16; off > 0; off >>= 1) s += __shfl_xor(s, off, 32);
    if (lane == 0) logits[row] = s;
}

// ---------------------------------------------------------------------------
// Kernel 4: per-batch softmax over S=4096 logits.
// ---------------------------------------------------------------------------
__global__ __launch_bounds__(1024)
void softmax_kernel(const float* __restrict__ logits, float* __restrict__ p) {
    __shared__ float red[1024];
    const int b = blockIdx.x;
    const int t = threadIdx.x;
    const float* lg = logits + (size_t)b * SEQ;

    float v[4];
    float mx = -3.402823466e+38f;
#pragma unroll
    for (int j = 0; j < 4; ++j) { v[j] = lg[t + 1024 * j]; mx = fmaxf(mx, v[j]); }
    red[t] = mx; __syncthreads();
    for (int s = 512; s > 0; s >>= 1) { if (t < s) red[t] = fmaxf(red[t], red[t + s]); __syncthreads(); }
    const float m = red[0]; __syncthreads();

    float sum = 0.0f;
#pragma unroll
    for (int j = 0; j < 4; ++j) { v[j] = __expf(v[j] - m); sum += v[j]; }
    red[t] = sum; __syncthreads();
    for (int s = 512; s > 0; s >>= 1) { if (t < s) red[t] += red[t + s]; __syncthreads(); }
    const float inv = 1.0f / red[0];   // one precise divide per block: keep exact
#pragma unroll
    for (int j = 0; j < 4; ++j) p[(size_t)b * SEQ + t + 1024 * j] = v[j] * inv;
}

// ---------------------------------------------------------------------------
// Kernel 5/6: context[b,c] = sum_s p[b,s]*h[b,s,c]  (chunked partials, no atomics)
// ---------------------------------------------------------------------------
__global__ __launch_bounds__(512)
void context_partial_kernel(const float* __restrict__ h, const float* __restrict__ p,
                            float* __restrict__ part) {
    const int b = blockIdx.x >> 4;
    const int chunk = blockIdx.x & 15;
    const int c = threadIdx.x;
    float acc = 0.0f;
    const int s0 = chunk * 256;
    for (int s = s0; s < s0 + 256; ++s)
        acc += p[(size_t)b * SEQ + s] * h[((size_t)b * SEQ + s) * (2 * HDIM) + c];
    part[((size_t)b * 16 + chunk) * (2 * HDIM) + c] = acc;
}

__global__ __launch_bounds__(512)
void context_reduce_kernel(const float* __restrict__ part, float* __restrict__ ctx) {
    const int b = blockIdx.x;
    const int c = threadIdx.x;
    float s = 0.0f;
#pragma unroll
    for (int k = 0; k < 16; ++k) s += part[((size_t)b * 16 + k) * (2 * HDIM) + c];
    ctx[(size_t)b * (2 * HDIM) + c] = s;
}

// ---------------------------------------------------------------------------
// Kernel 7: out[b] = ctx[b,:] . out_w + out_b
// ---------------------------------------------------------------------------
__global__ __launch_bounds__(512)
void out_kernel(const float* __restrict__ ctx, const float* __restrict__ ow,
                const float* __restrict__ ob, float* __restrict__ out) {
    __shared__ float red[512];
    const int b = blockIdx.x;
    const int t = threadIdx.x;
    red[t] = ctx[(size_t)b * 512 + t] * ow[t];
    __syncthreads();
    for (int s = 256; s > 0; s >>= 1) { if (t < s) red[t] += red[t + s]; __syncthreads(); }
    if (t == 0) out[b] = red[0] + ob[0];
}

// ---------------------------------------------------------------------------
extern "C" void kernel_launch(void* const* d_in, const int* in_sizes, int n_in,
                              void* d_out, int out_size, void* d_ws, size_t ws_size,
                              hipStream_t stream) {
    const int*   X   = (const int*)  d_in[0];
    const float* emb = (const float*)d_in[1];
    const float* wf  = (const float*)d_in[2];
    const float* bfv = (const float*)d_in[3];
    const float* wb  = (const float*)d_in[4];
    const float* bbv = (const float*)d_in[5];
    const float* Mu  = (const float*)d_in[6];
    const float* ow  = (const float*)d_in[7];
    const float* ob  = (const float*)d_in[8];
    float* out = (float*)d_out;

    // workspace layout (bytes)
    char* ws = (char*)d_ws;
    const size_t GATE_BYTES = (size_t)2 * NTOK * HDIM * sizeof(float);   // 256 MB
    float* gateZ  = (float*)(ws);                                        // 256 MB
    float* gateF  = (float*)(ws + GATE_BYTES);                           // 256 MB
    float* h      = (float*)(ws + 2 * GATE_BYTES);                       // 256 MB
    float* logits = (float*)(ws + 2 * GATE_BYTES + (size_t)NTOK * 2 * HDIM * sizeof(float));
    float* p      = (float*)((char*)logits + (size_t)NTOK * sizeof(float));
    float* part   = (float*)((char*)p      + (size_t)NTOK * sizeof(float));
    float* ctx    = (float*)((char*)part   + (size_t)BATCH * 16 * 2 * HDIM * sizeof(float));

    gemm_gates_kernel<<<NTOK / 64, 256, 0, stream>>>(X, emb, wf, bfv, wb, bbv, gateZ, gateF);
    scan_kernel<<<16384 / 256, 256, 0, stream>>>(gateZ, gateF, h);
    logits_kernel<<<NTOK / 8, 256, 0, stream>>>(h, Mu, logits);
    softmax_kernel<<<BATCH, 1024, 0, stream>>>(logits, p);
    context_partial_kernel<<<BATCH * 16, 512, 0, stream>>>(h, p, part);
    context_reduce_kernel<<<BATCH, 512, 0, stream>>>(part, ctx);
    out_kernel<<<BATCH, 512, 0, stream>>>(ctx, ow, ob, out);
    (void)in_sizes; (void)n_in; (void)out_size; (void)ws_size;
}